// MQADecoder_20787641712998
// MI455X (gfx1250) — compile-verified
//
#include <hip/hip_runtime.h>
#include <hip/hip_bf16.h>

// ---------------- problem constants ----------------
constexpr int B  = 32;
constexpr int M  = 4096;
constexpr int H  = 32;
constexpr int D  = 4096;
constexpr int KD = 128;
constexpr int VD = 128;
constexpr int MP1 = M + 1;

constexpr int DSPLIT = 4;    // D-split for projection kernel
constexpr int MSPLIT = 16;   // M-split for flash attention
constexpr int ROWS_PER_SPLIT = (MP1 + MSPLIT - 1) / MSPLIT;  // 257

typedef float v2f __attribute__((ext_vector_type(2)));
typedef float v8f __attribute__((ext_vector_type(8)));

#define NEG_INF (-__builtin_inff())

__device__ __forceinline__ v8f wmma_f32(v2f a, v2f b, v8f c) {
  // D = A(16x4 f32) * B(4x16 f32) + C(16x16 f32)
  return __builtin_amdgcn_wmma_f32_16x16x4_f32(false, a, false, b, (short)0, c,
                                               false, false);
}

// ---- CDNA5 async global->LDS staging (ASYNCcnt path, no VGPR round-trip) ----
// IOFFSET is added to BOTH the LDS and the global address (ISA 08 §4.4), so a
// single (lds, global) base pair serves several consecutive b128 transfers.
__device__ __forceinline__ unsigned lds_off(const void* p) {
  // generic -> group addrspace cast is a truncation on AMDGPU
  return (unsigned)(unsigned long long)p;
}

__device__ __forceinline__ void async_copy64(unsigned lds, const void* g) {
  asm volatile(
      "global_load_async_to_lds_b128 %0, %1, off\n\t"
      "global_load_async_to_lds_b128 %0, %1, off offset:16\n\t"
      "global_load_async_to_lds_b128 %0, %1, off offset:32\n\t"
      "global_load_async_to_lds_b128 %0, %1, off offset:48" ::"v"(lds),
      "v"(g)
      : "memory");
}

__device__ __forceinline__ void async_copy128(unsigned lds, const void* g) {
  asm volatile(
      "global_load_async_to_lds_b128 %0, %1, off\n\t"
      "global_load_async_to_lds_b128 %0, %1, off offset:16\n\t"
      "global_load_async_to_lds_b128 %0, %1, off offset:32\n\t"
      "global_load_async_to_lds_b128 %0, %1, off offset:48\n\t"
      "global_load_async_to_lds_b128 %0, %1, off offset:64\n\t"
      "global_load_async_to_lds_b128 %0, %1, off offset:80\n\t"
      "global_load_async_to_lds_b128 %0, %1, off offset:96\n\t"
      "global_load_async_to_lds_b128 %0, %1, off offset:112" ::"v"(lds),
      "v"(g)
      : "memory");
}

__device__ __forceinline__ void wait_async0() {
  asm volatile("s_wait_asynccnt 0x0" ::: "memory");
}

// =====================================================================
// Kernel 1: bulk copy of the KV cache into the output concat buffers.
// Pure streaming copy; also warms L2 (134MB fits in 192MB) for attention.
// =====================================================================
__global__ __launch_bounds__(256) void copy_kernel(
    const float4* __restrict__ Kp, const float4* __restrict__ Vp,
    float4* __restrict__ Kn, float4* __restrict__ Vn) {
  const long total  = (long)B * M * KD / 4;   // float4s per tensor
  const long rowq   = (long)M * KD / 4;       // float4s per src batch
  const long dstrow = (long)MP1 * KD / 4;     // float4s per dst batch
  const long stride = (long)gridDim.x * blockDim.x;
  for (long i = (long)blockIdx.x * blockDim.x + threadIdx.x; i < 2 * total;
       i += stride) {
    const bool isV = (i >= total);
    const long t   = isV ? i - total : i;
    const long b   = t / rowq;
    const long r   = t - b * rowq;
    if (!isV) Kn[b * dstrow + r] = Kp[t];
    else      Vn[b * dstrow + r] = Vp[t];
  }
}

// =====================================================================
// Kernel 2: projections  q = x*P_q[h],  k_new = x*P_k,  v_new = x*P_v
// grid = (DSPLIT, H+2); each block: 32x128 output tile, D/DSPLIT contraction.
// =====================================================================
__global__ __launch_bounds__(256) void proj_kernel(
    const float* __restrict__ x, const float* __restrict__ Pq,
    const float* __restrict__ Pk, const float* __restrict__ Pv,
    float* __restrict__ q_part, float* __restrict__ kv_part) {
  const int s    = blockIdx.x;        // D split
  const int u    = blockIdx.y;        // 0..H-1: head, H: P_k, H+1: P_v
  const int lane = threadIdx.x & 31;
  const int wave = threadIdx.x >> 5;
  const int l16  = lane & 15;
  const bool hi  = lane >= 16;

  __shared__ float xs[32][128];       // x chunk: [batch][d_local]

  const float* Wbase;
  if (u < H)       Wbase = Pq + (size_t)u * D * KD;
  else if (u == H) Wbase = Pk;
  else             Wbase = Pv;

  const int rtile = wave >> 2;        // batch row tile 0..1
  const int ct0   = (wave & 3) * 2;   // two col tiles ct0, ct0+1

  v8f acc0 = {0, 0, 0, 0, 0, 0, 0, 0};
  v8f acc1 = {0, 0, 0, 0, 0, 0, 0, 0};

  const int dBeg = s * (D / DSPLIT);
  const int dEnd = dBeg + (D / DSPLIT);

  const int xr  = threadIdx.x >> 3;
  const int xqd = (threadIdx.x & 7) * 16;
  const unsigned xs_lds = lds_off(&xs[xr][xqd]);

  for (int dc = dBeg; dc < dEnd; dc += 128) {
    // async-stage x[:, dc:dc+128] (32x128 f32) straight into LDS
    async_copy64(xs_lds, x + (size_t)xr * D + dc + xqd);
    wait_async0();
    __syncthreads();

    const int arow = rtile * 16 + l16;
    for (int kk = 0; kk < 128; kk += 4) {
      const int ac = kk + (hi ? 2 : 0);
      const v2f a = *(const v2f*)(&xs[arow][ac]);
      const int brow = dc + ac;
      const float* w0 = Wbase + (size_t)brow * KD + ct0 * 16 + l16;
      v2f b0; b0.x = w0[0];  b0.y = w0[KD];
      v2f b1; b1.x = w0[16]; b1.y = w0[16 + KD];
      acc0 = wmma_f32(a, b0, acc0);
      acc1 = wmma_f32(a, b1, acc1);
    }
    __syncthreads();
  }

  // write partial C tiles to workspace
  for (int j = 0; j < 8; ++j) {
    const int row = rtile * 16 + j + (hi ? 8 : 0);   // batch
    const int c0  = ct0 * 16 + l16;                  // k column
    if (u < H) {
      float* dst = q_part + (size_t)s * B * H * KD + ((size_t)row * H + u) * KD;
      dst[c0]      = acc0[j];
      dst[c0 + 16] = acc1[j];
    } else {
      float* dst =
          kv_part + ((size_t)s * 2 + (u - H)) * B * KD + (size_t)row * KD;
      dst[c0]      = acc0[j];
      dst[c0 + 16] = acc1[j];
    }
  }
}

// =====================================================================
// Kernel 3: reduce D-split partials.  q[b,h,k] = sum_s q_part.
// Also writes row M of K_new / V_new (the appended token).
// =====================================================================
__global__ __launch_bounds__(256) void reduce_kernel(
    const float* __restrict__ q_part, const float* __restrict__ kv_part,
    float* __restrict__ q, float* __restrict__ Kn, float* __restrict__ Vn) {
  const int NQ  = B * H * KD;  // 131072
  const int idx = blockIdx.x * 256 + threadIdx.x;
  if (idx < NQ) {
    float s = 0.f;
    for (int p = 0; p < DSPLIT; ++p) s += q_part[(size_t)p * NQ + idx];
    q[idx] = s;
  } else {
    const int t    = idx - NQ;        // 0..8191
    const int half = t >> 12;         // 0 = K, 1 = V
    const int r    = t & 4095;        // b*128 + k
    float s = 0.f;
    for (int p = 0; p < DSPLIT; ++p)
      s += kv_part[((size_t)p * 2 + half) * B * KD + r];
    const int b = r >> 7, k = r & 127;
    float* dst = (half == 0) ? Kn : Vn;
    dst[((size_t)b * MP1 + M) * KD + k] = s;
  }
}

// =====================================================================
// Kernel 4: flash attention over the shared cache, split over M.
// K/V chunks are streamed with GLOBAL_LOAD_ASYNC_TO_LDS_B128 (ASYNCcnt).
// =====================================================================
__global__ __launch_bounds__(256) void attn_kernel(
    const float* __restrict__ q, const float* __restrict__ Kn,
    const float* __restrict__ Vn, float* __restrict__ o_part,
    float* __restrict__ m_part, float* __restrict__ l_part) {
  const int b     = blockIdx.x;
  const int split = blockIdx.y;
  const int lane  = threadIdx.x & 31;
  const int wave  = threadIdx.x >> 5;
  const int l16   = lane & 15;
  const bool hi   = lane >= 16;

  __shared__ float qs[32][128];   // q[b]  : [h][k]
  __shared__ float Ks[32][128];   // chunk : [m_local][k]
  __shared__ float Vs[32][128];   // chunk : [m_local][v]
  __shared__ float Ss[32][32];    // logits then probs : [h][m_local]
  __shared__ float scale_s[32];   // per-row rescale for this chunk

  // async-stage q[b] (32x128) into LDS
  async_copy64(lds_off(((float*)&qs[0][0]) + threadIdx.x * 16),
               q + (size_t)b * H * KD + (size_t)threadIdx.x * 16);

  float run_m = NEG_INF, run_l = 0.f;  // live in lanes of threads 0..31

  const int hr  = wave >> 2;           // output h row-tile (stage 3)
  const int vt0 = (wave & 3) * 2;      // two v col-tiles
  v8f acc0 = {0, 0, 0, 0, 0, 0, 0, 0};
  v8f acc1 = {0, 0, 0, 0, 0, 0, 0, 0};

  const int mStart = split * ROWS_PER_SPLIT;
  int mEnd = mStart + ROWS_PER_SPLIT;
  if (mEnd > MP1) mEnd = MP1;

  const size_t kvbase = (size_t)b * MP1 * KD;

  const int sr  = threadIdx.x >> 3;          // staging row 0..31
  const int sqd = (threadIdx.x & 7) * 16;    // staging col offset
  const unsigned ks_lds = lds_off(&Ks[sr][sqd]);
  const unsigned vs_lds = lds_off(&Vs[sr][sqd]);

  for (int mc = mStart; mc < mEnd; mc += 32) {
    {  // async-stage K/V chunk (rows clamped; invalid masked in softmax)
      int mrow = mc + sr;
      if (mrow > M) mrow = M;
      async_copy64(ks_lds, Kn + kvbase + (size_t)mrow * KD + sqd);
      async_copy64(vs_lds, Vn + kvbase + (size_t)mrow * VD + sqd);
    }
    wait_async0();
    __syncthreads();

    // ---- stage 1: S = q * K^T  (4 tiles of 16x16, waves 0..3) ----
    if (wave < 4) {
      const int shr = wave >> 1, smt = wave & 1;
      v8f sacc = {0, 0, 0, 0, 0, 0, 0, 0};
      const int hrow = shr * 16 + l16;
      const int mloc = smt * 16 + l16;
      for (int kk = 0; kk < 128; kk += 4) {
        const int c = kk + (hi ? 2 : 0);
        const v2f a  = *(const v2f*)&qs[hrow][c];
        const v2f bb = *(const v2f*)&Ks[mloc][c];
        sacc = wmma_f32(a, bb, sacc);
      }
      for (int j = 0; j < 8; ++j)
        Ss[shr * 16 + j + (hi ? 8 : 0)][smt * 16 + l16] = sacc[j];
    }
    __syncthreads();

    // ---- stage 2: online softmax bookkeeping (one thread per h row) ----
    if (threadIdx.x < 32) {
      const int h = threadIdx.x;
      const int valid = (mEnd - mc < 32) ? (mEnd - mc) : 32;
      float cm = NEG_INF;
      for (int i = 0; i < valid; ++i) cm = fmaxf(cm, Ss[h][i]);
      const float nm = fmaxf(run_m, cm);
      const float sc = __expf(run_m - nm);
      float ps = 0.f;
      for (int i = 0; i < 32; ++i) {
        const float p = (i < valid) ? __expf(Ss[h][i] - nm) : 0.f;
        Ss[h][i] = p;
        ps += p;
      }
      run_l = run_l * sc + ps;
      run_m = nm;
      scale_s[h] = sc;
    }
    __syncthreads();

    // ---- stage 3: o = o*scale + P * V  (all 8 waves, 2 tiles each) ----
    for (int j = 0; j < 8; ++j) {
      const float f = scale_s[hr * 16 + j + (hi ? 8 : 0)];
      acc0[j] *= f;
      acc1[j] *= f;
    }
    {
      const int hrow = hr * 16 + l16;
      const int col0 = vt0 * 16 + l16;
      for (int kk = 0; kk < 32; kk += 4) {
        const int c = kk + (hi ? 2 : 0);
        const v2f a = *(const v2f*)&Ss[hrow][c];
        v2f b0; b0.x = Vs[c][col0];      b0.y = Vs[c + 1][col0];
        v2f b1; b1.x = Vs[c][col0 + 16]; b1.y = Vs[c + 1][col0 + 16];
        acc0 = wmma_f32(a, b0, acc0);
        acc1 = wmma_f32(a, b1, acc1);
      }
    }
    __syncthreads();
  }

  // ---- write raw o partial + stats ----
  const size_t obase = ((size_t)b * MSPLIT + split) * H * VD;
  for (int j = 0; j < 8; ++j) {
    const int h = hr * 16 + j + (hi ? 8 : 0);
    const int v = vt0 * 16 + l16;
    o_part[obase + (size_t)h * VD + v]      = acc0[j];
    o_part[obase + (size_t)h * VD + v + 16] = acc1[j];
  }
  if (threadIdx.x < 32) {
    const size_t sidx = ((size_t)b * MSPLIT + split) * H + threadIdx.x;
    m_part[sidx] = run_m;
    l_part[sidx] = run_l;
  }
}

// =====================================================================
// Kernel 5: merge M-splits:  o = sum_s e^(m_s - mmax) o_s / sum_s e^... l_s
// =====================================================================
__global__ __launch_bounds__(128) void combine_kernel(
    const float* __restrict__ o_part, const float* __restrict__ m_part,
    const float* __restrict__ l_part, float* __restrict__ o) {
  const int bh = blockIdx.x;
  const int b = bh / H, h = bh % H;
  const int v = threadIdx.x;
  float mmax = NEG_INF;
  for (int s = 0; s < MSPLIT; ++s)
    mmax = fmaxf(mmax, m_part[((size_t)b * MSPLIT + s) * H + h]);
  float num = 0.f, den = 0.f;
  for (int s = 0; s < MSPLIT; ++s) {
    const size_t sidx = ((size_t)b * MSPLIT + s) * H + h;
    const float w = __expf(m_part[sidx] - mmax);
    den += w * l_part[sidx];
    num += w * o_part[(((size_t)b * MSPLIT + s) * H + h) * VD + v];
  }
  o[((size_t)b * H + h) * VD + v] = num / den;
}

// =====================================================================
// Kernel 6: y[b,d] = sum_{h,v} o[b,h,v] * P_o[h,d,v]
// P_o tiles async-staged into padded LDS (stride 132 floats).
// =====================================================================
__global__ __launch_bounds__(256) void oproj_kernel(
    const float* __restrict__ o, const float* __restrict__ Po,
    float* __restrict__ y) {
  const int d0   = blockIdx.x * 64;
  const int lane = threadIdx.x & 31;
  const int wave = threadIdx.x >> 5;
  const int l16  = lane & 15;
  const bool hi  = lane >= 16;

  __shared__ float oh[32][128];   // o[:, h, :]   : [batch][v]
  __shared__ float T[64][132];    // P_o[h, d0+dl, v] : [dl][v], padded

  const int rt = wave >> 2;       // batch row tile 0..1
  const int ct = wave & 3;        // d col tile 0..3
  v8f acc = {0, 0, 0, 0, 0, 0, 0, 0};

  const int orr = threadIdx.x >> 3;
  const int oqd = (threadIdx.x & 7) * 16;
  const unsigned oh_lds = lds_off(&oh[orr][oqd]);
  const int dl  = threadIdx.x >> 2;
  const int seg = (threadIdx.x & 3) * 32;
  const unsigned T_lds = lds_off(&T[dl][seg]);

  for (int h = 0; h < H; ++h) {
    // async-stage o[:, h, :] (32x128) and P_o[h, d0:d0+64, :] (64x128)
    async_copy64(oh_lds, o + ((size_t)orr * H + h) * VD + oqd);
    async_copy128(T_lds, Po + ((size_t)h * D + d0 + dl) * VD + seg);
    wait_async0();
    __syncthreads();

    const int brow = rt * 16 + l16;
    const int drow = ct * 16 + l16;
    for (int vv = 0; vv < 128; vv += 4) {
      const int c = vv + (hi ? 2 : 0);
      const v2f a  = *(const v2f*)&oh[brow][c];
      const v2f bb = *(const v2f*)&T[drow][c];
      acc = wmma_f32(a, bb, acc);
    }
    __syncthreads();
  }
  for (int j = 0; j < 8; ++j) {
    const int row = rt * 16 + j + (hi ? 8 : 0);
    y[(size_t)row * D + d0 + ct * 16 + l16] = acc[j];
  }
}

// =====================================================================
extern "C" void kernel_launch(void* const* d_in, const int* in_sizes, int n_in,
                              void* d_out, int out_size, void* d_ws,
                              size_t ws_size, hipStream_t stream) {
  const float* x      = (const float*)d_in[0];
  const float* K_prev = (const float*)d_in[1];
  const float* V_prev = (const float*)d_in[2];
  const float* P_q    = (const float*)d_in[3];
  const float* P_k    = (const float*)d_in[4];
  const float* P_v    = (const float*)d_in[5];
  const float* P_o    = (const float*)d_in[6];

  float* y  = (float*)d_out;
  float* Kn = y + (size_t)B * D;                 // [B][MP1][KD]
  float* Vn = Kn + (size_t)B * MP1 * KD;         // [B][MP1][VD]

  // workspace carve-up (floats)
  float* ws      = (float*)d_ws;
  float* q_part  = ws;                                         // DSPLIT*B*H*KD
  float* kv_part = q_part + (size_t)DSPLIT * B * H * KD;       // DSPLIT*2*B*KD
  float* qbuf    = kv_part + (size_t)DSPLIT * 2 * B * KD;      // B*H*KD
  float* o_part  = qbuf + (size_t)B * H * KD;                  // MSPLIT*B*H*VD
  float* m_part  = o_part + (size_t)MSPLIT * B * H * VD;       // MSPLIT*B*H
  float* l_part  = m_part + (size_t)MSPLIT * B * H;            // MSPLIT*B*H
  float* obuf    = l_part + (size_t)MSPLIT * B * H;            // B*H*VD

  // 1) cache copy (warms L2 for attention)
  copy_kernel<<<4096, 256, 0, stream>>>((const float4*)K_prev,
                                        (const float4*)V_prev, (float4*)Kn,
                                        (float4*)Vn);
  // 2) projections (D-split partials)
  proj_kernel<<<dim3(DSPLIT, H + 2), 256, 0, stream>>>(x, P_q, P_k, P_v,
                                                       q_part, kv_part);
  // 3) reduce partials; append k_new/v_new at row M
  reduce_kernel<<<(B * H * KD + 2 * B * KD) / 256, 256, 0, stream>>>(
      q_part, kv_part, qbuf, Kn, Vn);
  // 4) flash attention, split over M
  attn_kernel<<<dim3(B, MSPLIT), 256, 0, stream>>>(qbuf, Kn, Vn, o_part,
                                                   m_part, l_part);
  // 5) merge splits
  combine_kernel<<<B * H, 128, 0, stream>>>(o_part, m_part, l_part, obuf);
  // 6) output projection
  oproj_kernel<<<D / 64, 256, 0, stream>>>(obuf, P_o, y);
}